// GCN_19619410608308
// MI455X (gfx1250) — compile-verified
//
#include <hip/hip_runtime.h>
#include <hip/hip_bf16.h>

typedef float v2f __attribute__((ext_vector_type(2)));
typedef float v8f __attribute__((ext_vector_type(8)));

#define IN_F 256
#define HID   64
#define NCLS  16

// ---------------- utility: zero a float buffer ----------------
__global__ void gcn_zero_kernel(float* __restrict__ p, int n) {
    int i = blockIdx.x * blockDim.x + threadIdx.x;
    if (i < n) p[i] = 0.0f;
}

// ---------------- degree accumulation (fp32 hardware atomics) ----------------
__global__ void gcn_degree_kernel(const int* __restrict__ src, const int* __restrict__ dst,
                                  float* __restrict__ deg_out, float* __restrict__ deg_in, int e) {
    int i = blockIdx.x * blockDim.x + threadIdx.x;
    if (i < e) {
        unsafeAtomicAdd(&deg_out[src[i]], 1.0f);
        unsafeAtomicAdd(&deg_in[dst[i]], 1.0f);
    }
}

// deg -> rsqrt(max(deg,1))  (norm_src and norm_dst are contiguous: length 2N)
__global__ void gcn_norm_kernel(float* __restrict__ deg, int n2) {
    int i = blockIdx.x * blockDim.x + threadIdx.x;
    if (i < n2) deg[i] = rsqrtf(fmaxf(deg[i], 1.0f));
}

// ---------------- fp32 WMMA GEMM: C[M x NC] = A[M x K] * B[K x NC] ----------------
// One wave computes a full 16 x NC row stripe (TN = NC/16 accumulators), so each
// A fragment is loaded ONCE per K-step and reused for TN WMMAs (X read exactly once).
// M must be a multiple of 16 (50000 = 3125*16).  K, NC are compile-time for
// 32-bit shift-based addressing and full unrolling.
// Fragment layouts per CDNA5 ISA 7.12.2:
//   A 16x4 f32 : lanes 0-15 -> M=lane, K={0,1} in v0,v1 ; lanes 16-31 -> K={2,3}
//   B 4x16 f32 : lanes 0-15 -> N=lane, K={0,1} in v0,v1 ; lanes 16-31 -> K={2,3}
//   C 16x16    : VGPR j -> row j (lanes 0-15) / row j+8 (lanes 16-31), col = lane%16
template <int K, int NC>
__global__ void gcn_wmma_gemm_kernel(const float* __restrict__ A, const float* __restrict__ B,
                                     float* __restrict__ C, int M) {
    constexpr int TN = NC / 16;
    int w = (blockIdx.x * blockDim.x + threadIdx.x) >> 5;   // global wave id = row stripe
    if (w >= (M >> 4)) return;                               // wave-uniform: EXEC stays all-1s
    int lane = threadIdx.x & 31;
    int half = lane >> 4;            // 0: K pair {0,1}, 1: K pair {2,3}
    int lr   = lane & 15;            // row (A) / col (B,C) index within tile

    const float* arow = A + (w * 16 + lr) * K;               // A row for this lane (32-bit idx)

    v8f acc[TN];
#pragma unroll
    for (int t = 0; t < TN; ++t) acc[t] = (v8f){};

#pragma unroll 4
    for (int k = 0; k < K; k += 4) {
        int kk = k + 2 * half;
        v2f a = *(const v2f*)(arow + kk);                    // K, K+1 contiguous, 8B aligned
#pragma unroll
        for (int t = 0; t < TN; ++t) {
            v2f b;
            b.x = B[(kk)     * NC + t * 16 + lr];
            b.y = B[(kk + 1) * NC + t * 16 + lr];
            acc[t] = __builtin_amdgcn_wmma_f32_16x16x4_f32(
                /*neg_a=*/false, a, /*neg_b=*/false, b,
                /*c_mod=*/(short)0, acc[t], /*reuse_a=*/false, /*reuse_b=*/false);
        }
    }

    float* crow = C + (w * 16 + 8 * half) * NC + lr;
#pragma unroll
    for (int t = 0; t < TN; ++t)
#pragma unroll
        for (int j = 0; j < 8; ++j)
            crow[j * NC + t * 16] = acc[t][j];
}

// ---------------- layer-1 edge scatter: 16 threads/edge, float4 each ----------------
__global__ void gcn_edge64_kernel(const int* __restrict__ src, const int* __restrict__ dst,
                                  const float* __restrict__ h,        // [N,64]
                                  const float* __restrict__ norm_src, // [N]
                                  float* __restrict__ agg, int e) {
    int t = blockIdx.x * blockDim.x + threadIdx.x;
    if (t >= e * (HID / 4)) return;
    int ed = t >> 4;                 // HID/4 == 16 chunks per edge
    int c  = (t & 15) * 4;
    int s = src[ed], d = dst[ed];
    float ns = norm_src[s];
    float4 v = *(const float4*)(h + s * HID + c);
    float* a = agg + d * HID + c;
    unsafeAtomicAdd(a + 0, v.x * ns);
    unsafeAtomicAdd(a + 1, v.y * ns);
    unsafeAtomicAdd(a + 2, v.z * ns);
    unsafeAtomicAdd(a + 3, v.w * ns);
}

// ---------------- layer-2 edge scatter: 4 threads/edge, float4 each ----------------
__global__ void gcn_edge16_kernel(const int* __restrict__ src, const int* __restrict__ dst,
                                  const float* __restrict__ h,        // [N,16]
                                  const float* __restrict__ norm_src, // [N]
                                  float* __restrict__ agg, int e) {
    int t = blockIdx.x * blockDim.x + threadIdx.x;
    if (t >= e * (NCLS / 4)) return;
    int ed = t >> 2;
    int c  = (t & 3) * 4;
    int s = src[ed], d = dst[ed];
    float ns = norm_src[s];
    float4 v = *(const float4*)(h + s * NCLS + c);
    float* a = agg + d * NCLS + c;
    unsafeAtomicAdd(a + 0, v.x * ns);
    unsafeAtomicAdd(a + 1, v.y * ns);
    unsafeAtomicAdd(a + 2, v.z * ns);
    unsafeAtomicAdd(a + 3, v.w * ns);
}

// ---------------- fused epilogues ----------------
// agg1 = relu(agg1 * norm_dst + b1)   (in place, [N,64])
__global__ void gcn_bias_relu_kernel(float* __restrict__ agg, const float* __restrict__ norm_dst,
                                     const float* __restrict__ b, int total) {
    int i = blockIdx.x * blockDim.x + threadIdx.x;
    if (i < total) {
        int r = i >> 6;              // /HID
        int c = i & (HID - 1);
        agg[i] = fmaxf(fmaf(agg[i], norm_dst[r], b[c]), 0.0f);
    }
}

// out = out * norm_dst + b2   (in place, [N,16])
__global__ void gcn_final_kernel(float* __restrict__ out, const float* __restrict__ norm_dst,
                                 const float* __restrict__ b, int total) {
    int i = blockIdx.x * blockDim.x + threadIdx.x;
    if (i < total) {
        int r = i >> 4;              // /NCLS
        int c = i & (NCLS - 1);
        out[i] = fmaf(out[i], norm_dst[r], b[c]);
    }
}

extern "C" void kernel_launch(void* const* d_in, const int* in_sizes, int n_in,
                              void* d_out, int out_size, void* d_ws, size_t ws_size,
                              hipStream_t stream) {
    const float* X   = (const float*)d_in[0];
    const int*   src = (const int*)  d_in[1];
    const int*   dst = (const int*)  d_in[2];
    const float* W1  = (const float*)d_in[3];
    const float* b1  = (const float*)d_in[4];
    const float* W2  = (const float*)d_in[5];
    const float* b2  = (const float*)d_in[6];
    float* out = (float*)d_out;

    const int n = in_sizes[0] / IN_F;    // 50000
    const int e = in_sizes[1];           // 800000

    // workspace layout (floats): norm_src[n] | norm_dst[n] | h1[n*64] | agg1[n*64] | h2[n*16]
    float* ws       = (float*)d_ws;
    float* norm_src = ws;
    float* norm_dst = ws + n;
    float* h1       = ws + 2 * (size_t)n;
    float* agg1     = h1 + (size_t)n * HID;
    float* h2       = agg1 + (size_t)n * HID;

    const int B = 256;
    // 1) zero degree accumulators, agg1, and d_out
    gcn_zero_kernel<<<(2 * n + B - 1) / B, B, 0, stream>>>(norm_src, 2 * n);
    gcn_zero_kernel<<<(n * HID + B - 1) / B, B, 0, stream>>>(agg1, n * HID);
    gcn_zero_kernel<<<(n * NCLS + B - 1) / B, B, 0, stream>>>(out, n * NCLS);

    // 2) degrees, 3) norms
    gcn_degree_kernel<<<(e + B - 1) / B, B, 0, stream>>>(src, dst, norm_src, norm_dst, e);
    gcn_norm_kernel<<<(2 * n + B - 1) / B, B, 0, stream>>>(norm_src, 2 * n);

    // 4) h1 = X @ W1  (fp32 WMMA, one 16x64 row stripe per wave, 4 waves/block)
    {
        int waves = n / 16;                                  // 3125 row stripes
        gcn_wmma_gemm_kernel<IN_F, HID><<<(waves + 3) / 4, 128, 0, stream>>>(X, W1, h1, n);
    }

    // 5) layer-1 gather/scale/scatter-add
    gcn_edge64_kernel<<<(e * (HID / 4) + B - 1) / B, B, 0, stream>>>(src, dst, h1, norm_src, agg1, e);

    // 6) agg1 = relu(agg1 * norm_dst + b1)
    gcn_bias_relu_kernel<<<(n * HID + B - 1) / B, B, 0, stream>>>(agg1, norm_dst, b1, n * HID);

    // 7) h2 = agg1 @ W2  (fp32 WMMA, K=64, one 16x16 tile per wave)
    {
        int waves = n / 16;
        gcn_wmma_gemm_kernel<HID, NCLS><<<(waves + 3) / 4, 128, 0, stream>>>(agg1, W2, h2, n);
    }

    // 8) layer-2 gather/scale/scatter-add into d_out
    gcn_edge16_kernel<<<(e * (NCLS / 4) + B - 1) / B, B, 0, stream>>>(src, dst, h2, norm_src, out, e);

    // 9) out = out * norm_dst + b2
    gcn_final_kernel<<<(n * NCLS + B - 1) / B, B, 0, stream>>>(out, norm_dst, b2, n * NCLS);
}